// BaselineModel_53979148976382
// MI455X (gfx1250) — compile-verified
//
#include <hip/hip_runtime.h>
#include <math.h>

// ---------------------------------------------------------------------------
// Sparse voxel conv net, restructured as: per-layer fused
//   gather27(grid_prev at flat_prev[p]) -> 27x(3x3) matmul -> ReLU
//   -> atomic scatter into grid_next at flat_next[p]
// Head [N,3]@[3,20] done with V_WMMA_F32_16X16X4_F32 (fp32, exact shapes).
// ---------------------------------------------------------------------------

typedef float v2f __attribute__((ext_vector_type(2)));
typedef float v8f __attribute__((ext_vector_type(8)));

#define TPB 256

static inline size_t alignUp(size_t x, size_t a) { return (x + a - 1) / a * a; }

// ---- voxelize all three resolutions once ----------------------------------
__global__ void compute_flat_kernel(const float* __restrict__ coords, int n,
                                    int* __restrict__ f02, int* __restrict__ f04,
                                    int* __restrict__ f08) {
  int p = blockIdx.x * blockDim.x + threadIdx.x;
  if (p >= n) return;
  float x = coords[p];
  float y = coords[n + p];
  float z = coords[2 * n + p];
  {
    int vx = (int)floorf(x / 0.02f) + 1;
    int vy = (int)floorf(y / 0.02f) + 1;
    int vz = (int)floorf(z / 0.02f) + 1;
    f02[p] = (vx * 203 + vy) * 203 + vz;
  }
  {
    int vx = (int)floorf(x / 0.04f) + 1;
    int vy = (int)floorf(y / 0.04f) + 1;
    int vz = (int)floorf(z / 0.04f) + 1;
    f04[p] = (vx * 103 + vy) * 103 + vz;
  }
  {
    int vx = (int)floorf(x / 0.08f) + 1;
    int vy = (int)floorf(y / 0.08f) + 1;
    int vz = (int)floorf(z / 0.08f) + 1;
    f08[p] = (vx * 53 + vy) * 53 + vz;
  }
}

// ---- initial scatter of pixel_vals into the fine grid ---------------------
__global__ void scatter_kernel(const float* __restrict__ feat,
                               const int* __restrict__ flat,
                               float* __restrict__ grid, int n) {
  int p = blockIdx.x * blockDim.x + threadIdx.x;
  if (p >= n) return;
  const float* f = feat + (size_t)p * 3;
  float* g = grid + (size_t)flat[p] * 3;
  atomicAdd(&g[0], f[0]);
  atomicAdd(&g[1], f[1]);
  atomicAdd(&g[2], f[2]);
}

// ---- fused 27-tap conv (+bias,ReLU) then either scatter-to-next-grid or
//      write per-point features (for the head). flip=1 reverses kernel taps.
__global__ void conv_kernel(const float* __restrict__ gridPrev, int G,
                            const int* __restrict__ flatPrev,
                            const float* __restrict__ W, int flip,
                            const float* __restrict__ bias,
                            float* __restrict__ gridNext,
                            const int* __restrict__ flatNext,
                            float* __restrict__ featOut, int n) {
  __shared__ float Wlds[243];  // [27][Cin=3][Cout=3], tap-flipped if needed
  for (int i = threadIdx.x; i < 243; i += blockDim.x) {
    int k = i / 9, r = i - k * 9;
    int ks = flip ? (26 - k) : k;
    Wlds[i] = W[ks * 9 + r];
  }
  __syncthreads();

  int p = blockIdx.x * blockDim.x + threadIdx.x;
  if (p >= n) return;

  float a0 = 0.f, a1 = 0.f, a2 = 0.f;
  if (bias) { a0 = bias[0]; a1 = bias[1]; a2 = bias[2]; }

  int fp = flatPrev[p];
  int k = 0;
#pragma unroll
  for (int dx = -1; dx <= 1; ++dx)
#pragma unroll
    for (int dy = -1; dy <= 1; ++dy)
#pragma unroll
      for (int dz = -1; dz <= 1; ++dz) {
        int nb = fp + (dx * G + dy) * G + dz;
        const float* g = gridPrev + (size_t)nb * 3;
        float g0 = g[0], g1 = g[1], g2 = g[2];
        const float* w = &Wlds[k * 9];
        a0 = fmaf(g0, w[0], fmaf(g1, w[3], fmaf(g2, w[6], a0)));
        a1 = fmaf(g0, w[1], fmaf(g1, w[4], fmaf(g2, w[7], a1)));
        a2 = fmaf(g0, w[2], fmaf(g1, w[5], fmaf(g2, w[8], a2)));
        ++k;
      }

  a0 = fmaxf(a0, 0.f);
  a1 = fmaxf(a1, 0.f);
  a2 = fmaxf(a2, 0.f);

  if (featOut) {
    float* f = featOut + (size_t)p * 3;
    f[0] = a0; f[1] = a1; f[2] = a2;
  } else {
    float* gn = gridNext + (size_t)flatNext[p] * 3;
    atomicAdd(&gn[0], a0);
    atomicAdd(&gn[1], a1);
    atomicAdd(&gn[2], a2);
  }
}

// ---- head: out[N,20] = feat[N,3] @ w_gen[3,20] + b_gen via WMMA f32 16x16x4
// One wave handles 16 points; two WMMAs cover N columns 0..15 and 16..19.
// A layout (ISA 7.12.2, 32-bit A 16x4): lanes 0-15 M, v0/v1 = K0/K1;
// lanes 16-31 same M, v0/v1 = K2/K3.  B mirrors with lane = N.
__global__ void gen_wmma_kernel(const float* __restrict__ feat,
                                const float* __restrict__ wg,
                                const float* __restrict__ bg,
                                float* __restrict__ out, int n, int ntiles) {
  int wavesPerBlock = blockDim.x >> 5;
  int tile = blockIdx.x * wavesPerBlock + (threadIdx.x >> 5);
  if (tile >= ntiles) return;                 // wave-uniform: EXEC stays full
  int lane = threadIdx.x & 31;
  int half = lane >> 4;                       // 0 -> K{0,1}, 1 -> K{2,3}
  int m = lane & 15;
  int base = tile * 16;

  // A: 16 points x K(=3 real, K=3 padded to 0)
  float a0 = 0.f, a1 = 0.f;
  int pA = base + m;
  if (pA < n) {
    const float* fr = feat + (size_t)pA * 3;
    if (half) { a0 = fr[2]; /* K=3 pad */ }
    else      { a0 = fr[0]; a1 = fr[1]; }
  }
  v2f a; a.x = a0; a.y = a1;

  // B1: w_gen cols 0..15   B2: cols 16..19 (padded)
  int n1 = m;
  v2f b1, b2;
  b1.x = half ? wg[2 * 20 + n1] : wg[0 * 20 + n1];
  b1.y = half ? 0.f             : wg[1 * 20 + n1];
  bool c2 = n1 < 4;
  int n2 = 16 + n1;
  b2.x = c2 ? (half ? wg[2 * 20 + n2] : wg[0 * 20 + n2]) : 0.f;
  b2.y = (c2 && !half) ? wg[1 * 20 + n2] : 0.f;

  v8f c = {};
  v8f d1 = __builtin_amdgcn_wmma_f32_16x16x4_f32(false, a, false, b1,
                                                 (short)0, c, false, false);
  v8f d2 = __builtin_amdgcn_wmma_f32_16x16x4_f32(false, a, false, b2,
                                                 (short)0, c, false, false);

  float bias1 = bg[n1];
  float bias2 = c2 ? bg[n2] : 0.f;
#pragma unroll
  for (int v = 0; v < 8; ++v) {
    int row = v + half * 8;                   // D: lane 0-15 M=v, 16-31 M=v+8
    int p = base + row;
    if (p < n) {
      float* orow = out + (size_t)p * 20;
      orow[n1] = d1[v] + bias1;
      if (c2) orow[n2] = d2[v] + bias2;
    }
  }
}

// ---------------------------------------------------------------------------
extern "C" void kernel_launch(void* const* d_in, const int* in_sizes, int n_in,
                              void* d_out, int out_size, void* d_ws,
                              size_t ws_size, hipStream_t stream) {
  (void)n_in; (void)out_size; (void)ws_size;
  const float* coords     = (const float*)d_in[0];
  const float* pixel_vals = (const float*)d_in[1];
  const float* w_enc1 = (const float*)d_in[2];
  const float* w_enc2 = (const float*)d_in[3];
  const float* w_enc3 = (const float*)d_in[4];
  const float* w_dec1 = (const float*)d_in[5];
  const float* b_dec1 = (const float*)d_in[6];
  const float* w_dec2 = (const float*)d_in[7];
  const float* b_dec2 = (const float*)d_in[8];
  const float* w_dec3 = (const float*)d_in[9];
  const float* b_dec3 = (const float*)d_in[10];
  const float* w_gen  = (const float*)d_in[11];
  const float* b_gen  = (const float*)d_in[12];
  float* out = (float*)d_out;
  int n = in_sizes[0] / 3;

  const int G02 = 203, G04 = 103, G08 = 53;   // ceil(4/vs)+3
  const size_t szA = (size_t)G02 * G02 * G02 * 3 * sizeof(float);  // ~100 MB
  const size_t szB = (size_t)G04 * G04 * G04 * 3 * sizeof(float);  // ~13 MB
  const size_t szC = (size_t)G08 * G08 * G08 * 3 * sizeof(float);  // ~1.8 MB

  char* ws = (char*)d_ws;
  size_t off = 0;
  auto take = [&](size_t bytes) {
    size_t o = off; off = alignUp(off + bytes, 256); return o;
  };
  int*   f02  = (int*)(ws + take((size_t)n * sizeof(int)));
  int*   f04  = (int*)(ws + take((size_t)n * sizeof(int)));
  int*   f08  = (int*)(ws + take((size_t)n * sizeof(int)));
  float* feat = (float*)(ws + take((size_t)n * 3 * sizeof(float)));
  float* bufA = (float*)(ws + take(szA));   // vs=0.02 grids
  float* bufB = (float*)(ws + take(szB));   // vs=0.04 grids
  float* bufC1 = (float*)(ws + take(szC));  // vs=0.08 grid (enc3 input)
  float* bufC2 = (float*)(ws + take(szC));  // vs=0.08 grid (dec1 input)

  int nb = (n + TPB - 1) / TPB;

  compute_flat_kernel<<<nb, TPB, 0, stream>>>(coords, n, f02, f04, f08);

  // enc1 (vs=0.02): pool pixel_vals -> bufA
  hipMemsetAsync(bufA, 0, szA, stream);
  scatter_kernel<<<nb, TPB, 0, stream>>>(pixel_vals, f02, bufA, n);
  // enc1 conv -> pool into vs=0.04 grid (enc2 input)
  hipMemsetAsync(bufB, 0, szB, stream);
  conv_kernel<<<nb, TPB, 0, stream>>>(bufA, G02, f02, w_enc1, 0, nullptr,
                                      bufB, f04, nullptr, n);
  // enc2 conv -> pool into vs=0.08 grid (enc3 input)
  hipMemsetAsync(bufC1, 0, szC, stream);
  conv_kernel<<<nb, TPB, 0, stream>>>(bufB, G04, f04, w_enc2, 0, nullptr,
                                      bufC1, f08, nullptr, n);
  // enc3 conv -> pool into vs=0.08 grid (dec1 input)
  hipMemsetAsync(bufC2, 0, szC, stream);
  conv_kernel<<<nb, TPB, 0, stream>>>(bufC1, G08, f08, w_enc3, 0, nullptr,
                                      bufC2, f08, nullptr, n);
  // dec1 (transpose, vs=0.08) -> pool into vs=0.04 grid
  hipMemsetAsync(bufB, 0, szB, stream);
  conv_kernel<<<nb, TPB, 0, stream>>>(bufC2, G08, f08, w_dec1, 1, b_dec1,
                                      bufB, f04, nullptr, n);
  // dec2 (transpose, vs=0.04) -> pool into vs=0.02 grid
  hipMemsetAsync(bufA, 0, szA, stream);
  conv_kernel<<<nb, TPB, 0, stream>>>(bufB, G04, f04, w_dec2, 1, b_dec2,
                                      bufA, f02, nullptr, n);
  // dec3 (transpose, vs=0.02) -> per-point features for the head
  conv_kernel<<<nb, TPB, 0, stream>>>(bufA, G02, f02, w_dec3, 1, b_dec3,
                                      nullptr, nullptr, feat, n);

  // head: [N,3] @ [3,20] + b via WMMA f32 16x16x4, one wave per 16 points
  int ntiles = (n + 15) / 16;
  int wavesPerBlock = 8;  // 256 threads
  int genBlocks = (ntiles + wavesPerBlock - 1) / wavesPerBlock;
  gen_wmma_kernel<<<genBlocks, wavesPerBlock * 32, 0, stream>>>(
      feat, w_gen, b_gen, out, n, ntiles);
}